// Transformer_F_80118319940299
// MI455X (gfx1250) — compile-verified
//
#include <hip/hip_runtime.h>

// ---------------------------------------------------------------------------
// Transformer_F reference, MI455X (gfx1250, wave32) implementation.
//
// Per layer, per head:
//   U = B@Z, W = C@Z, key = V@Z            (32x32 @ 32x2049)
//   attn[l,j] = all-k sign-match of U[:,l] vs W[:,j]   (T x T, ~all zero)
//   out[:,j] += sum_{l<T-1, attn[l,j]} key[:,l]
//
// D == 32 == wave32: one column's sign pattern packs into one u32.
// Stage A: WMMA f32 16x16x4 projections -> keyT (transposed) + sign masks.
// Stage C: per-column ballot scan over packed masks; gathers key columns
//          only on (astronomically rare) matches. Exact in f32.
// ---------------------------------------------------------------------------

#define TT 2049
#define DD 32

typedef float v2f __attribute__((ext_vector_type(2)));
typedef float v8f __attribute__((ext_vector_type(8)));

// ---------------------------------------------------------------------------
// Stage A: one wave (32 threads) per (16-column tile, head*3+mat) pair.
//   mat 0 = V -> keyT[h][t][d] (transposed store, coalesced gathers later)
//   mat 1 = B -> posU/negU masks,  mat 2 = C -> posW/negW masks
// masks layout: masks[(h*4 + which)*TT + t], which: 0=posU 1=negU 2=posW 3=negW
// ---------------------------------------------------------------------------
__global__ __launch_bounds__(32) void stageA(const float* __restrict__ Zc,
                                             const float* __restrict__ param,   // layer base
                                             float* __restrict__ keyT,          // [2][TT][32]
                                             unsigned int* __restrict__ masks)  // [2][4][TT]
{
    __shared__ float lds[16 * 33];   // padded tile for column-major mask build

    const int jt   = blockIdx.x;       // 0..128 column tiles
    const int g    = blockIdx.y;       // h*3 + c
    const int h    = g / 3;
    const int c    = g % 3;
    const float* P = param + (size_t)(h * 3 + c) * (DD * DD);

    const int lane  = threadIdx.x;
    const int half  = lane >> 4;       // 0: K offset +0/+1, 1: K offset +2/+3
    const int m     = lane & 15;
    const int jbase = jt * 16;
    const int t     = jbase + m;
    const int tc    = (t < TT) ? t : (TT - 1);   // clamp last ragged tile

    v8f acc0 = {};   // rows 0..15
    v8f acc1 = {};   // rows 16..31

#pragma unroll
    for (int k0 = 0; k0 < DD; k0 += 4) {
        const int ka = k0 + 2 * half;
        v2f a0, a1, b;
        // A (16x4 f32): lane=M, VGPR0 K=ka, VGPR1 K=ka+1
        a0.x = P[m * DD + ka];            a0.y = P[m * DD + ka + 1];
        a1.x = P[(m + 16) * DD + ka];     a1.y = P[(m + 16) * DD + ka + 1];
        // B (4x16 f32): lane%16 = N column, half selects K row pair
        b.x  = Zc[ka * TT + tc];          b.y  = Zc[(ka + 1) * TT + tc];
        acc0 = __builtin_amdgcn_wmma_f32_16x16x4_f32(false, a0, false, b,
                                                     (short)0, acc0, false, false);
        acc1 = __builtin_amdgcn_wmma_f32_16x16x4_f32(false, a1, false, b,
                                                     (short)0, acc1, false, false);
    }

    if (c == 0) {
        // key = V@Z : store transposed keyT[h][t][d]
        if (t < TT) {
            float* kT = keyT + ((size_t)h * TT + (size_t)t) * DD;
#pragma unroll
            for (int r = 0; r < 8; ++r) {
                kT[8 * half + r]      = acc0[r];   // C layout: M = r + 8*half
                kT[16 + 8 * half + r] = acc1[r];
            }
        }
    } else {
        // U or W tile -> per-column 32-bit sign masks via LDS transpose
#pragma unroll
        for (int r = 0; r < 8; ++r) {
            lds[m * 33 + 8 * half + r]      = acc0[r];
            lds[m * 33 + 16 + 8 * half + r] = acc1[r];
        }
        __syncthreads();
        if (lane < 16) {
            unsigned int pos = 0u, neg = 0u;
#pragma unroll
            for (int d = 0; d < DD; ++d) {
                float v = lds[lane * 33 + d];
                pos |= (v > 0.0f ? 1u : 0u) << d;
                neg |= (v < 0.0f ? 1u : 0u) << d;
            }
            const int tt2 = jbase + lane;
            if (tt2 < TT) {
                const int which = (c == 1) ? 0 : 2;  // B -> U masks, C -> W masks
                unsigned int* mb = masks + ((size_t)h * 4 + which) * TT;
                mb[tt2]      = pos;
                mb[TT + tt2] = neg;
            }
        }
    }
}

// ---------------------------------------------------------------------------
// Stage C: Zout[:,j] = Zc[:,j] + sum_h sum_{l<2048, match} key_h[:,l]
// One wave per output column j; lane == feature d. U-masks staged in LDS
// and shared by the block's 8 waves. Ballot scan: 64 iters x 2 heads.
// ---------------------------------------------------------------------------
__global__ __launch_bounds__(256) void stageC(const float* __restrict__ Zc,
                                              const float* __restrict__ keyT,    // [2][TT][32]
                                              const unsigned int* __restrict__ masks,
                                              float* __restrict__ Zout)
{
    __shared__ unsigned int smask[2 * 2 * 2048];   // [h][pos/neg][l], 32 KB

    for (int i = threadIdx.x; i < 2 * 2 * 2048; i += 256) {
        const int h    = i >> 12;
        const int kind = (i >> 11) & 1;            // 0=posU 1=negU
        const int l    = i & 2047;
        smask[i] = masks[((size_t)h * 4 + kind) * TT + l];
    }
    __syncthreads();

    const int wid  = threadIdx.x >> 5;
    const int lane = threadIdx.x & 31;
    const int j    = blockIdx.x * 8 + wid;
    if (j >= TT) return;

    float acc = Zc[lane * TT + j];                 // residual base

#pragma unroll
    for (int h = 0; h < 2; ++h) {
        const unsigned int* mH = masks + (size_t)h * 4 * TT;
        const unsigned int pW  = mH[2 * TT + j];
        const unsigned int nW  = mH[3 * TT + j];
        const unsigned int* sU = smask + h * 4096;
        const float* kT        = keyT + (size_t)h * TT * DD;

        for (int l0 = 0; l0 < 2048; l0 += 32) {    // l = T-1 excluded by mask M
            const unsigned int pU = sU[l0 + lane];
            const unsigned int nU = sU[2048 + l0 + lane];
            const bool match = (((pU & pW) | (nU & nW)) == 0xFFFFFFFFu);
            unsigned int mm = __builtin_amdgcn_ballot_w32(match);
            while (mm) {                            // uniform, ~never taken
                const int l = l0 + __builtin_ctz(mm);
                acc += kT[l * DD + lane];           // coalesced 128B gather
                mm &= mm - 1;
            }
        }
    }
    Zout[lane * TT + j] = acc;
}

// ---------------------------------------------------------------------------
extern "C" void kernel_launch(void* const* d_in, const int* in_sizes, int n_in,
                              void* d_out, int out_size, void* d_ws, size_t ws_size,
                              hipStream_t stream)
{
    (void)in_sizes; (void)n_in; (void)out_size; (void)ws_size;

    const float* Z        = (const float*)d_in[0];   // (32, 2049) f32
    const float* allparam = (const float*)d_in[1];   // (2,2,3,32,32) f32
    float* out            = (float*)d_out;           // (32, 2049) f32

    char* ws = (char*)d_ws;
    const size_t zBytes   = (size_t)DD * TT * sizeof(float);            // 262272
    const size_t keyBytes = (size_t)2 * TT * DD * sizeof(float);        // 524544
    float*        Z1    = (float*)ws;
    float*        keyT  = (float*)(ws + ((zBytes + 255) & ~(size_t)255));
    unsigned int* masks = (unsigned int*)(ws + ((zBytes + 255) & ~(size_t)255)
                                             + ((keyBytes + 255) & ~(size_t)255));

    const dim3 gA(129, 6), bA(32);
    const dim3 gC((TT + 7) / 8), bC(256);
    const size_t layerStride = (size_t)2 * 3 * DD * DD;   // params per layer

    // Layer 0: Z -> Z1
    stageA<<<gA, bA, 0, stream>>>(Z, allparam, keyT, masks);
    stageC<<<gC, bC, 0, stream>>>(Z, keyT, masks, Z1);
    // Layer 1: Z1 -> out
    stageA<<<gA, bA, 0, stream>>>(Z1, allparam + layerStride, keyT, masks);
    stageC<<<gC, bC, 0, stream>>>(Z1, keyT, masks, out);
}